// ParametrizedInhibition_25022479467200
// MI455X (gfx1250) — compile-verified
//
#include <hip/hip_runtime.h>

typedef __attribute__((ext_vector_type(2))) float v2f;
typedef __attribute__((ext_vector_type(8))) float v8f;

#define C_DIM 512
#define HW    1024
#define KC    64
#define LDA   68    // LDS A row stride (floats): 272B, 16B aligned, bank-spread
#define LDB   132   // LDS B row stride (floats): 528B, 16B aligned, bank-spread
#define NTILE 128   // n (hw) tile per block
#define OTILE 64    // o tile per block

// ---------------------------------------------------------------------------
// Step 1: a[t] = first row of (I - T):  a[0]=1, a[d mod 512] = -ricker(d)
// Ricker offsets d in [-31,31], center tap zeroed (SELF_CONNECT=False).
// ---------------------------------------------------------------------------
__global__ __launch_bounds__(512) void k_build_a(const float* __restrict__ wp,
                                                 const float* __restrict__ dp,
                                                 float* __restrict__ a) {
  int t = (int)threadIdx.x;                 // 0..511
  float width = wp[0], damp = dp[0];
  int d = (t < 256) ? t : t - 512;          // ring offset
  float val = (t == 0) ? 1.0f : 0.0f;
  int ad = d < 0 ? -d : d;
  if (ad >= 1 && ad <= 31) {
    float x   = (float)d;
    float amp = 2.0f / (sqrtf(3.0f * width) * 1.3313353638003897f); // pi^0.25
    float xw  = x / width;
    float r   = damp * amp * (1.0f - xw * xw) *
                __expf(-(x * x) / (2.0f * width * width));
    val -= r;
  }
  a[t] = val;
}

// ---------------------------------------------------------------------------
// Step 2: eigenvalues of the circulant (a real & even -> DFT is real):
//   lam[k] = sum_j a[j] * cos(2*pi*j*k/512)
// Integer-mod phase -> exact argument reduction.
// ---------------------------------------------------------------------------
__global__ __launch_bounds__(512) void k_dft(const float* __restrict__ a,
                                             float* __restrict__ lam) {
  __shared__ float sa[C_DIM];
  int k = (int)threadIdx.x;
  sa[k] = a[k];
  __syncthreads();
  const float w0 = 6.283185307179586f / 512.0f;
  float s = 0.0f;
  for (int j = 0; j < C_DIM; ++j) {
    int t = (j * k) & 511;
    s += sa[j] * __cosf(w0 * (float)t);
  }
  lam[k] = s;
}

// ---------------------------------------------------------------------------
// Step 3: first row of inv(I - T):
//   invrow[i] = (1/512) * sum_k cos(2*pi*i*k/512) / lam[k]
// ---------------------------------------------------------------------------
__global__ __launch_bounds__(512) void k_invrow(const float* __restrict__ lam,
                                                float* __restrict__ invrow) {
  __shared__ float sl[C_DIM];
  int i = (int)threadIdx.x;
  sl[i] = 1.0f / lam[i];
  __syncthreads();
  const float w0 = 6.283185307179586f / 512.0f;
  float s = 0.0f;
  for (int k = 0; k < C_DIM; ++k) {
    int t = (i * k) & 511;
    s += sl[k] * __cosf(w0 * (float)t);
  }
  invrow[i] = s * (1.0f / 512.0f);
}

// ---------------------------------------------------------------------------
// Step 4: materialize M[o][c] = invrow[(c - o) mod 512]   (row-major 512x512)
// ---------------------------------------------------------------------------
__global__ __launch_bounds__(256) void k_build_m(const float* __restrict__ invrow,
                                                 float* __restrict__ M) {
  int idx = (int)(blockIdx.x * 256 + threadIdx.x);   // 0..512*512-1
  int o = idx >> 9;
  int c = idx & 511;
  M[idx] = invrow[(c - o) & 511];
}

// ---------------------------------------------------------------------------
// Step 5: out[b,o,hw] = sum_c M[o,c] * act[b,c,hw]
// Block: one batch, 64(o) x 128(hw) tile; 8 waves; wave w -> o sub-tile
// (w&3)*16, hw half (w>>2)*64 with FOUR 16x16 fp32 accumulators (4x A reuse).
// K staged through LDS in chunks of 64, consumed 4 at a time by
// v_wmma_f32_16x16x4_f32. Next stage is prefetched (global_prefetch_b8).
// ---------------------------------------------------------------------------
__global__ __launch_bounds__(256) void k_gemm(const float* __restrict__ M,
                                              const float* __restrict__ act,
                                              float* __restrict__ out) {
  __shared__ __align__(16) float As[OTILE][LDA];   // [o][k]
  __shared__ __align__(16) float Bs[KC][LDB];      // [k][n]

  const int nb = (int)blockIdx.x;       // hw tile: 0..7
  const int ob = (int)blockIdx.y;       // o  tile: 0..7
  const int b  = (int)blockIdx.z;       // batch
  const int o_base = ob * OTILE;
  const int n_base = nb * NTILE;

  const float* __restrict__ Xb = act + (size_t)b * C_DIM * HW;
  float* __restrict__ Ob       = out + (size_t)b * C_DIM * HW;

  const int tid  = (int)threadIdx.x;
  const int lane = tid & 31;
  const int wave = tid >> 5;            // 0..7
  const int wo   = wave & 3;            // o sub-tile (16 rows)
  const int wn   = wave >> 2;           // hw half    (64 cols)

  const int rl    = lane & 15;          // row-in-frag / col-in-frag
  const int half  = lane >> 4;          // 0 or 1
  const int khalf = half * 2;           // K offset inside A/B fragments

  v8f acc0 = {}, acc1 = {}, acc2 = {}, acc3 = {};

  for (int kc = 0; kc < C_DIM; kc += KC) {
    // ---- stage A (64 o-rows x 64 k) ----
    #pragma unroll
    for (int q = tid; q < (OTILE * KC) / 4; q += 256) {     // 1024 float4
      int row = q >> 4;
      int c4  = (q & 15) << 2;
      *(float4*)&As[row][c4] =
          *(const float4*)&M[(size_t)(o_base + row) * C_DIM + kc + c4];
    }
    // ---- stage B (64 k-rows x 128 n) ----
    #pragma unroll
    for (int q = tid; q < (KC * NTILE) / 4; q += 256) {     // 2048 float4
      int row = q >> 5;
      int c4  = (q & 31) << 2;
      *(float4*)&Bs[row][c4] =
          *(const float4*)&Xb[(size_t)(kc + row) * HW + n_base + c4];
    }

    // ---- prefetch next stage into cache while this stage computes ----
    if (kc + KC < C_DIM) {
      int pr = tid >> 2;                                    // 0..63
      __builtin_prefetch(&M[(size_t)(o_base + pr) * C_DIM + kc + KC], 0, 3);
      __builtin_prefetch(&Xb[(size_t)(kc + KC + pr) * HW + n_base], 0, 3);
      __builtin_prefetch(&Xb[(size_t)(kc + KC + pr) * HW + n_base + 64], 0, 3);
    }
    __syncthreads();

    // ---- consume: 16 K-steps of 4, four WMMAs per step (A reused 4x) ----
    #pragma unroll
    for (int k = 0; k < KC; k += 4) {
      // A fragment 16x4: lane rl holds row M=rl; VGPRs = K {khalf, khalf+1}
      v2f afrag = *(const v2f*)&As[wo * 16 + rl][k + khalf];
      // B fragments 4x16: lane rl holds col N=rl; VGPRs = K {khalf, khalf+1}
      const int cb = wn * 64 + rl;
      v2f b0, b1, b2, b3;
      b0.x = Bs[k + khalf][cb];          b0.y = Bs[k + khalf + 1][cb];
      b1.x = Bs[k + khalf][cb + 16];     b1.y = Bs[k + khalf + 1][cb + 16];
      b2.x = Bs[k + khalf][cb + 32];     b2.y = Bs[k + khalf + 1][cb + 32];
      b3.x = Bs[k + khalf][cb + 48];     b3.y = Bs[k + khalf + 1][cb + 48];

      acc0 = __builtin_amdgcn_wmma_f32_16x16x4_f32(
          false, afrag, false, b0, (short)0, acc0, false, false);
      acc1 = __builtin_amdgcn_wmma_f32_16x16x4_f32(
          false, afrag, false, b1, (short)0, acc1, false, false);
      acc2 = __builtin_amdgcn_wmma_f32_16x16x4_f32(
          false, afrag, false, b2, (short)0, acc2, false, false);
      acc3 = __builtin_amdgcn_wmma_f32_16x16x4_f32(
          false, afrag, false, b3, (short)0, acc3, false, false);
    }
    __syncthreads();
  }

  // ---- store: C/D layout: VGPR v -> row v + 8*half, col rl ----
  const int row0 = o_base + wo * 16 + half * 8;
  const int col0 = n_base + wn * 64 + rl;
  #pragma unroll
  for (int v = 0; v < 8; ++v) {
    float* r = &Ob[(size_t)(row0 + v) * HW + col0];
    r[0]  = acc0[v];
    r[16] = acc1[v];
    r[32] = acc2[v];
    r[48] = acc3[v];
  }
}

// ---------------------------------------------------------------------------
extern "C" void kernel_launch(void* const* d_in, const int* in_sizes, int n_in,
                              void* d_out, int out_size, void* d_ws, size_t ws_size,
                              hipStream_t stream) {
  const float* act = (const float*)d_in[0];   // [B, 512, 32, 32] fp32
  const float* wp  = (const float*)d_in[1];   // width scalar
  const float* dp  = (const float*)d_in[2];   // damp scalar
  float* out = (float*)d_out;
  float* ws  = (float*)d_ws;

  float* Mmat   = ws;                         // 512*512 floats (1 MB)
  float* avec   = ws + C_DIM * C_DIM;         // 512
  float* lam    = avec + C_DIM;               // 512
  float* invrow = lam + C_DIM;                // 512

  const int B = in_sizes[0] / (C_DIM * HW);   // 64

  k_build_a<<<1, 512, 0, stream>>>(wp, dp, avec);
  k_dft    <<<1, 512, 0, stream>>>(avec, lam);
  k_invrow <<<1, 512, 0, stream>>>(lam, invrow);
  k_build_m<<<(C_DIM * C_DIM) / 256, 256, 0, stream>>>(invrow, Mmat);

  dim3 grid(HW / NTILE, C_DIM / OTILE, B);    // (8, 8, 64)
  k_gemm<<<grid, 256, 0, stream>>>(Mmat, act, out);
}